// Inpainting_70540542869901
// MI455X (gfx1250) — compile-verified
//
#include <hip/hip_runtime.h>

// ===========================================================================
// MI455X / gfx1250 implementation of the wavelet-attention transformer block.
// Shapes: B=4, C=64, H=W=128, N=16384, HEADS=4 (16 ch/head), HID=170.
//
// All GEMM-like math (1x1 convs, Q.K^T, A.V) runs on V_WMMA_F32_16X16X4_F32
// (wave32, one wave owns a 16x16 f32 tile, K-loop step 4). f32 WMMA is chosen
// deliberately: the op is HBM-bandwidth bound (~0.5 GB traffic, AI <= ~21
// FLOP/B on the largest GEMM), so low-precision WMMA buys nothing while the
// reference relies on f32 accumulation (K=16384 attention reduction, LN).
//
// Round-2 changes:
//  * QK K-loop driven by a scalar trip count -> uniform s_cbranch, EXEC never
//    masked around WMMA (ISA: EXEC must be all-ones), less SALU overhead,
//    unroll 4 (4 WMMAs/iter).
//  * out_self*out_cross fused into the w_proj GEMM B-load (saves 32MB traffic).
//  * __builtin_prefetch (-> global_prefetch_b8) ahead of the strided row reads.
//
// Workspace plan (floats), peak ~176 MB:
//   [smalls 8192] [A 16MB] [Q 16MB] [X 16MB] [BIG: wavelets | KV | FFN reused]
// ===========================================================================

typedef __attribute__((ext_vector_type(2))) float v2f;
typedef __attribute__((ext_vector_type(8))) float v8f;

__device__ __forceinline__ v8f wmma_f32_16x16x4(v2f a, v2f b, v8f c) {
  // D = A(16x4) * B(4x16) + C, full f32.
  return __builtin_amdgcn_wmma_f32_16x16x4_f32(false, a, false, b,
                                               (short)0, c, false, false);
}

// db3 filters. Forward = reversed decomposition filters (jax conv is
// correlation); inverse (dilated correlation) uses the dec filters directly.
__constant__ float c_fwd_lo[6] = { 0.3326705529509569f, 0.8068915093133388f,
  0.4598775021193313f, -0.13501102001039084f, -0.08544127388224149f,
  0.035226291882100656f };
__constant__ float c_fwd_hi[6] = { 0.035226291882100656f, 0.08544127388224149f,
  -0.13501102001039084f, -0.4598775021193313f, 0.8068915093133388f,
  -0.3326705529509569f };
__constant__ float c_inv_lo[6] = { 0.035226291882100656f, -0.08544127388224149f,
  -0.13501102001039084f, 0.4598775021193313f, 0.8068915093133388f,
  0.3326705529509569f };
__constant__ float c_inv_hi[6] = { -0.3326705529509569f, 0.8068915093133388f,
  -0.4598775021193313f, -0.13501102001039084f, 0.08544127388224149f,
  0.035226291882100656f };

// --------------------------------------------------------------------------
// Channel LayerNorm over C of NCHW, eps=1e-5. One thread per (b, pixel).
// --------------------------------------------------------------------------
__global__ void k_chan_ln(const float* __restrict__ x, const float* __restrict__ gam,
                          const float* __restrict__ bet, float* __restrict__ out,
                          int B, int C, int N) {
  int idx = blockIdx.x * blockDim.x + threadIdx.x;
  if (idx >= B * N) return;
  int b = idx / N, n = idx % N;
  const float* p = x + (size_t)b * C * N + n;
  float s = 0.f, ss = 0.f;
  for (int c = 0; c < C; ++c) { float v = p[(size_t)c * N]; s += v; ss += v * v; }
  float m = s / (float)C;
  float var = ss / (float)C - m * m;
  float inv = rsqrtf(var + 1e-5f);
  float* q = out + (size_t)b * C * N + n;
  for (int c = 0; c < C; ++c) {
    float v = p[(size_t)c * N];
    q[(size_t)c * N] = (v - m) * inv * gam[c] + bet[c];
  }
}

// --------------------------------------------------------------------------
// Forward DWT along W: stride 2, zero-pad 4, taps 6. in (BC,H,W) -> (BC,H,Wo)
// --------------------------------------------------------------------------
__global__ void k_dwt_w(const float* __restrict__ x, float* __restrict__ lo,
                        float* __restrict__ hi, int BC, int H, int W, int Wo) {
  int idx = blockIdx.x * blockDim.x + threadIdx.x;
  int total = BC * H * Wo;
  if (idx >= total) return;
  int xo = idx % Wo; int t1 = idx / Wo; int y = t1 % H; int bc = t1 / H;
  const float* row = x + ((size_t)bc * H + y) * W;
  float alo = 0.f, ahi = 0.f;
#pragma unroll
  for (int t = 0; t < 6; ++t) {
    int xi = 2 * xo + t - 4;
    float v = (xi >= 0 && xi < W) ? row[xi] : 0.f;
    alo += v * c_fwd_lo[t];
    ahi += v * c_fwd_hi[t];
  }
  lo[idx] = alo; hi[idx] = ahi;
}

// Forward DWT along H: in (BC,Hi,W) -> (BC,Ho,W)
__global__ void k_dwt_h(const float* __restrict__ x, float* __restrict__ lo,
                        float* __restrict__ hi, int BC, int Hi, int W, int Ho) {
  int idx = blockIdx.x * blockDim.x + threadIdx.x;
  int total = BC * Ho * W;
  if (idx >= total) return;
  int xx = idx % W; int t1 = idx / W; int yo = t1 % Ho; int bc = t1 / Ho;
  const float* col = x + (size_t)bc * Hi * W + xx;
  float alo = 0.f, ahi = 0.f;
#pragma unroll
  for (int t = 0; t < 6; ++t) {
    int yi = 2 * yo + t - 4;
    float v = (yi >= 0 && yi < Hi) ? col[(size_t)yi * W] : 0.f;
    alo += v * c_fwd_lo[t];
    ahi += v * c_fwd_hi[t];
  }
  lo[idx] = alo; hi[idx] = ahi;
}

// Inverse (dilated) DWT along H: a*lo + b*hi. (BC,Hi,W) -> (BC,Ho,W), Ho=2*Hi-4
__global__ void k_idwt_h(const float* __restrict__ a, const float* __restrict__ b,
                         float* __restrict__ out, int BC, int Hi, int W, int Ho) {
  int idx = blockIdx.x * blockDim.x + threadIdx.x;
  int total = BC * Ho * W;
  if (idx >= total) return;
  int xx = idx % W; int t1 = idx / W; int i = t1 % Ho; int bc = t1 / Ho;
  const float* pa = a + (size_t)bc * Hi * W + xx;
  const float* pb = b + (size_t)bc * Hi * W + xx;
  float acc = 0.f;
#pragma unroll
  for (int t = 0; t < 6; ++t) {
    int p = i + t;
    if (p & 1) {
      int j = (p - 1) >> 1;
      if (j < Hi)
        acc += pa[(size_t)j * W] * c_inv_lo[t] + pb[(size_t)j * W] * c_inv_hi[t];
    }
  }
  out[idx] = acc;
}

// Inverse DWT along W. (BC,H,Wi) -> (BC,H,Wo)
__global__ void k_idwt_w(const float* __restrict__ a, const float* __restrict__ b,
                         float* __restrict__ out, int BC, int H, int Wi, int Wo) {
  int idx = blockIdx.x * blockDim.x + threadIdx.x;
  int total = BC * H * Wo;
  if (idx >= total) return;
  int i = idx % Wo; int t1 = idx / Wo; int y = t1 % H; int bc = t1 / H;
  const float* pa = a + ((size_t)bc * H + y) * Wi;
  const float* pb = b + ((size_t)bc * H + y) * Wi;
  float acc = 0.f;
#pragma unroll
  for (int t = 0; t < 6; ++t) {
    int p = i + t;
    if (p & 1) {
      int j = (p - 1) >> 1;
      if (j < Wi) acc += pa[j] * c_inv_lo[t] + pb[j] * c_inv_hi[t];
    }
  }
  out[idx] = acc;
}

// --------------------------------------------------------------------------
// Depthwise 3x3, pad 1 (correlation). x (B,C,H,W), w (C,1,3,3).
// --------------------------------------------------------------------------
__global__ void k_dw3(const float* __restrict__ x, const float* __restrict__ w,
                      float* __restrict__ out, int C, int H, int W, int total) {
  int idx = blockIdx.x * blockDim.x + threadIdx.x;
  if (idx >= total) return;
  int xx = idx % W; int t1 = idx / W; int y = t1 % H; int t2 = t1 / H;
  int c = t2 % C;
  const float* base = x + (size_t)t2 * H * W;
  const float* wk = w + (size_t)c * 9;
  float acc = 0.f;
#pragma unroll
  for (int ky = 0; ky < 3; ++ky) {
    int yi = y + ky - 1;
    if (yi < 0 || yi >= H) continue;
#pragma unroll
    for (int kx = 0; kx < 3; ++kx) {
      int xi = xx + kx - 1;
      if (xi < 0 || xi >= W) continue;
      acc += base[(size_t)yi * W + xi] * wk[ky * 3 + kx];
    }
  }
  out[idx] = acc;
}

// Fused depthwise 3x3 on 2*HID channels + GELU(exact) gating -> HID channels.
__global__ void k_dw3_gate(const float* __restrict__ x, const float* __restrict__ w,
                           float* __restrict__ out, int HID, int H, int W, int total) {
  int idx = blockIdx.x * blockDim.x + threadIdx.x;
  if (idx >= total) return;
  int xx = idx % W; int t1 = idx / W; int y = t1 % H; int t2 = t1 / H;
  int c = t2 % HID; int b = t2 / HID;
  const float* base1 = x + ((size_t)b * 2 * HID + c) * H * W;
  const float* base2 = base1 + (size_t)HID * H * W;
  const float* w1 = w + (size_t)c * 9;
  const float* w2 = w + (size_t)(c + HID) * 9;
  float a1 = 0.f, a2 = 0.f;
#pragma unroll
  for (int ky = 0; ky < 3; ++ky) {
    int yi = y + ky - 1;
    if (yi < 0 || yi >= H) continue;
#pragma unroll
    for (int kx = 0; kx < 3; ++kx) {
      int xi = xx + kx - 1;
      if (xi < 0 || xi >= W) continue;
      float s = base1[(size_t)yi * W + xi];
      float t = base2[(size_t)yi * W + xi];
      a1 += s * w1[ky * 3 + kx];
      a2 += t * w2[ky * 3 + kx];
    }
  }
  float g = 0.5f * a1 * (1.f + erff(a1 * 0.70710678118654752f));
  out[idx] = g * a2;
}

// --------------------------------------------------------------------------
// Row L2 norms: rows are (b, c) of a (B, Ctot, N) tensor, c in [Coff,Coff+Cuse)
// --------------------------------------------------------------------------
__global__ void k_rownorm(const float* __restrict__ x, float* __restrict__ norms,
                          int Ctot, int Coff, int Cuse, int N) {
  int r = blockIdx.x;
  int b = r / Cuse, c = r % Cuse;
  const float* p = x + ((size_t)(b * Ctot + Coff + c)) * N;
  float s = 0.f;
  for (int i = threadIdx.x; i < N; i += blockDim.x) { float v = p[i]; s += v * v; }
  __shared__ float sm[256];
  sm[threadIdx.x] = s;
  __syncthreads();
  for (int o = 128; o > 0; o >>= 1) {
    if (threadIdx.x < (unsigned)o) sm[threadIdx.x] += sm[threadIdx.x + o];
    __syncthreads();
  }
  if (threadIdx.x == 0) norms[r] = sqrtf(sm[0]);
}

// --------------------------------------------------------------------------
// Gram: scores[bh][c][d] = sum_n q[b,h*16+c,n] * k[b,kOff+h*16+d,n].
// 8 waves split K=16384; each accumulates a 16x16 f32 tile with WMMA
// 16x16x4 (K-step 4), partials reduced through LDS. The loop trip count is a
// scalar (uniform s_cbranch) so EXEC is never masked around the WMMAs.
// --------------------------------------------------------------------------
__global__ void k_qk_wmma(const float* __restrict__ q, int qCtot,
                          const float* __restrict__ kk, int kCtot, int kOff,
                          float* __restrict__ scores, int HD, int N) {
  int bh = blockIdx.x;
  int b = bh / HD, h = bh % HD;
  int lane = threadIdx.x & 31;
  int wv = threadIdx.x >> 5;
  int r = lane & 15;
  int koff = (lane >> 4) << 1;
  const float* qrow = q + ((size_t)(b * qCtot + h * 16 + r)) * N + koff;
  const float* krow = kk + ((size_t)(b * kCtot + kOff + h * 16 + r)) * N + koff;
  v8f acc = {};
  int iters = N >> 5;               // 8 waves * 4 K per iteration
  int base = wv << 2;
#pragma unroll 4
  for (int it = 0; it < iters; ++it) {
    int k0 = base + (it << 5);
    __builtin_prefetch(qrow + k0 + 64, 0, 1);   // global_prefetch_b8, 2 lines ahead
    __builtin_prefetch(krow + k0 + 64, 0, 1);
    v2f A;  A.x  = qrow[k0];  A.y  = qrow[k0 + 1];
    v2f Bv; Bv.x = krow[k0];  Bv.y = krow[k0 + 1];
    acc = wmma_f32_16x16x4(A, Bv, acc);
  }
  __shared__ float sm[8 * 256];
#pragma unroll
  for (int j = 0; j < 8; ++j) sm[wv * 256 + lane * 8 + j] = acc[j];
  __syncthreads();
  int t = threadIdx.x;
  float sum = 0.f;
#pragma unroll
  for (int w = 0; w < 8; ++w) sum += sm[w * 256 + t];
  int l = t >> 3, j = t & 7;
  int m = j + ((l >> 4) << 3);   // D layout: VGPR j -> M=j (+8 for lanes>=16)
  int n = l & 15;
  scores[((size_t)bh * 16 + m) * 16 + n] = sum;
}

// --------------------------------------------------------------------------
// Softmax over d of scores[bh][c][d] * temp[h] / (|q_c| * |k_d|); in place.
// One thread per row; rows = B*HD*16.
// --------------------------------------------------------------------------
__global__ void k_softmax16(float* __restrict__ sc, const float* __restrict__ qn,
                            const float* __restrict__ kn, const float* __restrict__ temp,
                            int HD, int rows, int useQn) {
  int t = blockIdx.x * blockDim.x + threadIdx.x;
  if (t >= rows) return;
  int c = t & 15;
  int bh = t >> 4;
  int h = bh % HD, b = bh / HD;
  float* row = sc + (size_t)t * 16;
  float qd = useQn ? fmaxf(qn[(b * HD + h) * 16 + c], 1e-12f) : 1.f;
  float tp = temp[h];
  float v[16]; float mx = -3.0e38f;
#pragma unroll
  for (int d = 0; d < 16; ++d) {
    float kd = fmaxf(kn[(b * HD + h) * 16 + d], 1e-12f);
    float s = row[d] / (qd * kd) * tp;
    v[d] = s; mx = fmaxf(mx, s);
  }
  float sum = 0.f;
#pragma unroll
  for (int d = 0; d < 16; ++d) { v[d] = expf(v[d] - mx); sum += v[d]; }
  float inv = 1.f / sum;
#pragma unroll
  for (int d = 0; d < 16; ++d) row[d] = v[d] * inv;
}

// --------------------------------------------------------------------------
// out[b, h*16+m, n0+n] = sum_k a[bh][m][k] * v[b, vOff+h*16+k, n0+n]
// One wave per (bh, 16-pixel tile); 4 WMMA 16x16x4 steps cover K=16.
// --------------------------------------------------------------------------
__global__ void k_av_wmma(const float* __restrict__ a, const float* __restrict__ v,
                          int vCtot, int vOff, float* __restrict__ out,
                          int HD, int N) {
  int NT = N >> 4;
  int nt = blockIdx.x % NT;
  int bh = blockIdx.x / NT;
  int b = bh / HD, h = bh % HD;
  int lane = threadIdx.x & 31;
  int r = lane & 15;
  int koff = (lane >> 4) << 1;
  int n0 = nt << 4;
  const float* abase = a + (size_t)bh * 256;
  v8f acc = {};
#pragma unroll
  for (int i = 0; i < 4; ++i) {
    int kb = i * 4 + koff;
    v2f A; A.x = abase[r * 16 + kb]; A.y = abase[r * 16 + kb + 1];
    const float* vp = v + ((size_t)(b * vCtot + vOff + h * 16 + kb)) * N + n0 + r;
    v2f Bv; Bv.x = vp[0]; Bv.y = vp[(size_t)N];
    acc = wmma_f32_16x16x4(A, Bv, acc);
  }
#pragma unroll
  for (int j = 0; j < 8; ++j) {
    int m = j + ((lane >> 4) << 3);
    out[((size_t)(b * (HD * 16) + h * 16 + m)) * N + n0 + r] = acc[j];
  }
}

// --------------------------------------------------------------------------
// 1x1 conv as GEMM: out[b,co,n] = sum_ci w[co,ci] * x[b,ci,n] * x2[b,ci,n]
// (+res). x2 optional (fused elementwise gate: out_self*out_cross). One wave
// per (b, 16-co tile, 16-pixel tile); K-loop step 4 with edge guards
// (handles Cin=170, Cout=340). Scalar K-loop -> EXEC untouched around WMMA.
// --------------------------------------------------------------------------
__global__ void k_gemm1x1_wmma(const float* __restrict__ x, const float* __restrict__ x2,
                               const float* __restrict__ w,
                               float* __restrict__ out, const float* __restrict__ res,
                               int Cin, int Cout, int CoT, int N) {
  int NT = N >> 4;
  int nt = blockIdx.x % NT;
  int t1 = blockIdx.x / NT;
  int cot = t1 % CoT;
  int b = t1 / CoT;
  int lane = threadIdx.x & 31;
  int r = lane & 15;
  int koff = (lane >> 4) << 1;
  int n0 = nt << 4;
  int co0 = cot << 4;
  int coA = co0 + r;
  const float* xb = x + (size_t)b * Cin * N + n0 + r;
  const float* x2b = x2 ? (x2 + (size_t)b * Cin * N + n0 + r) : nullptr;
  v8f acc = {};
  for (int k0 = 0; k0 < Cin; k0 += 4) {
    if (k0 + 8 < Cin) __builtin_prefetch(xb + (size_t)(k0 + 8) * N, 0, 1);
    int ka = k0 + koff, kb = ka + 1;
    float a0 = (coA < Cout && ka < Cin) ? w[(size_t)coA * Cin + ka] : 0.f;
    float a1 = (coA < Cout && kb < Cin) ? w[(size_t)coA * Cin + kb] : 0.f;
    float b0 = (ka < Cin) ? xb[(size_t)ka * N] : 0.f;
    float b1 = (kb < Cin) ? xb[(size_t)kb * N] : 0.f;
    if (x2b) {
      b0 *= (ka < Cin) ? x2b[(size_t)ka * N] : 0.f;
      b1 *= (kb < Cin) ? x2b[(size_t)kb * N] : 0.f;
    }
    v2f A; A.x = a0; A.y = a1;
    v2f Bv; Bv.x = b0; Bv.y = b1;
    acc = wmma_f32_16x16x4(A, Bv, acc);
  }
#pragma unroll
  for (int j = 0; j < 8; ++j) {
    int m = co0 + j + ((lane >> 4) << 3);
    if (m < Cout) {
      size_t oi = ((size_t)b * Cout + m) * N + n0 + r;
      float val = acc[j];
      if (res) val += res[oi];
      out[oi] = val;
    }
  }
}

// ===========================================================================
extern "C" void kernel_launch(void* const* d_in, const int* in_sizes, int n_in,
                              void* d_out, int out_size, void* d_ws, size_t ws_size,
                              hipStream_t stream) {
  const int B = 4, C = 64, H = 128, W = 128, N = H * W, HD = 4, HID = 170;
  const int SB = 66;            // db3 subband size for 128 with zero padding
  const int BC = B * C;

  const float* rgb    = (const float*)d_in[0];
  const float* depth  = (const float*)d_in[1];
  const float* temp   = (const float*)d_in[2];
  const float* w_qr   = (const float*)d_in[3];   // (128,64)
  const float* w_qd   = (const float*)d_in[4];   // (128,64)
  const float* w_qc   = (const float*)d_in[5];   // (128,1,3,3)
  const float* w_duck = (const float*)d_in[6];
  const float* w_c5   = (const float*)d_in[7];
  const float* w_c7   = (const float*)d_in[8];
  const float* w_c9   = (const float*)d_in[9];
  const float* w_proj = (const float*)d_in[10];  // (64,64)
  const float* ln1_g  = (const float*)d_in[11];
  const float* ln1_b  = (const float*)d_in[12];
  const float* ln2_g  = (const float*)d_in[13];
  const float* ln2_b  = (const float*)d_in[14];
  const float* w_fin  = (const float*)d_in[15];  // (340,64)
  const float* w_fdw  = (const float*)d_in[16];  // (340,1,3,3)
  const float* w_fout = (const float*)d_in[17];  // (64,170)
  float* y = (float*)d_out;

  // ---- workspace layout (floats) ----
  float* ws = (float*)d_ws;
  const size_t szA = (size_t)B * C * N;               // 4,194,304
  float* NQ  = ws + 0;                                // 256
  float* NK  = ws + 256;                              // 256
  float* SC  = ws + 512;                              // 4096 (B*HD*16*16)
  float* A   = ws + 8192;                             // norm_rgb / out_cross / x_ln
  float* Q   = A + szA;                               // q_rgb / x
  float* X   = Q + szA;                               // out_self
  float* BIG = X + szA;
  // wavelet phase
  const size_t szLW = (size_t)BC * H * SB;            // 2,162,688
  const size_t szSB = (size_t)BC * SB * SB;           // 1,115,136
  float* loW = BIG;            float* hiW = loW + szLW;
  float* ll  = hiW + szLW;     float* lh  = ll + szSB;
  float* hl  = lh + szSB;      float* hh  = hl + szSB;
  float* ll2 = hh + szSB;      float* lh2 = ll2 + szSB;
  float* hl2 = lh2 + szSB;     float* hh2 = hl2 + szSB;
  float* loR = loW;            float* hiR = hiW;      // reuse after consumed
  // KV phase (reuses BIG after wavelets done)
  const size_t szKV = (size_t)B * 2 * C * N;          // 8,388,608
  float* KVa = BIG;            float* KVb = BIG + szKV;
  // FFN phase (reuses BIG after attention done)
  const size_t szF1 = (size_t)B * 2 * HID * N;        // 22,282,240
  float* F1 = BIG;             float* G = BIG + szF1; // gated (B,HID,N)
  (void)ws_size; (void)in_sizes; (void)n_in; (void)out_size;

  const int TB = 256;
  auto blocks = [](long long n, int tb) { return (unsigned)((n + tb - 1) / tb); };

  // ---- 1) norm_rgb = LN(rgb) ----
  k_chan_ln<<<blocks((long long)B * N, TB), TB, 0, stream>>>(rgb, ln1_g, ln1_b, A, B, C, N);

  // ---- 2) MFE(norm_rgb) -> q_rgb in Q ----
  k_dwt_w<<<blocks((long long)BC * H * SB, TB), TB, 0, stream>>>(A, loW, hiW, BC, H, W, SB);
  k_dwt_h<<<blocks((long long)BC * SB * SB, TB), TB, 0, stream>>>(loW, ll, lh, BC, H, SB, SB);
  k_dwt_h<<<blocks((long long)BC * SB * SB, TB), TB, 0, stream>>>(hiW, hl, hh, BC, H, SB, SB);
  k_dw3<<<blocks((long long)BC * SB * SB, TB), TB, 0, stream>>>(ll, w_duck, ll2, C, SB, SB, BC * SB * SB);
  k_dw3<<<blocks((long long)BC * SB * SB, TB), TB, 0, stream>>>(lh, w_c5, lh2, C, SB, SB, BC * SB * SB);
  k_dw3<<<blocks((long long)BC * SB * SB, TB), TB, 0, stream>>>(hl, w_c7, hl2, C, SB, SB, BC * SB * SB);
  k_dw3<<<blocks((long long)BC * SB * SB, TB), TB, 0, stream>>>(hh, w_c9, hh2, C, SB, SB, BC * SB * SB);
  k_idwt_h<<<blocks((long long)BC * H * SB, TB), TB, 0, stream>>>(ll2, lh2, loR, BC, SB, SB, H);
  k_idwt_h<<<blocks((long long)BC * H * SB, TB), TB, 0, stream>>>(hl2, hh2, hiR, BC, SB, SB, H);
  k_idwt_w<<<blocks((long long)BC * H * W, TB), TB, 0, stream>>>(loR, hiR, Q, BC, H, SB, W);

  // ---- 3) k_r, v_r = split(dw3(conv1x1(norm_rgb, w_qkv_rgb))) ----
  k_gemm1x1_wmma<<<(unsigned)(B * 8 * (N / 16)), 32, 0, stream>>>(A, nullptr, w_qr, KVa, nullptr, C, 2 * C, 8, N);
  k_dw3<<<blocks((long long)B * 2 * C * N, TB), TB, 0, stream>>>(KVa, w_qc, KVb, 2 * C, H, W, B * 2 * C * N);

  // ---- 4) attention 1 (self) -> out_self in X ----
  k_rownorm<<<B * C, 256, 0, stream>>>(Q, NQ, C, 0, C, N);        // |q_c|
  k_rownorm<<<B * C, 256, 0, stream>>>(KVb, NK, 2 * C, 0, C, N);  // |k_d|
  k_qk_wmma<<<B * HD, 256, 0, stream>>>(Q, C, KVb, 2 * C, 0, SC, HD, N);
  k_softmax16<<<1, 256, 0, stream>>>(SC, NQ, NK, temp, HD, B * HD * 16, 1);
  k_av_wmma<<<(unsigned)(B * HD * (N / 16)), 32, 0, stream>>>(SC, KVb, 2 * C, C, X, HD, N);

  // ---- 5) attention 2 (cross, depth) -> out_cross in A ----
  k_gemm1x1_wmma<<<(unsigned)(B * 8 * (N / 16)), 32, 0, stream>>>(depth, nullptr, w_qd, KVa, nullptr, C, 2 * C, 8, N);
  k_dw3<<<blocks((long long)B * 2 * C * N, TB), TB, 0, stream>>>(KVa, w_qc, KVb, 2 * C, H, W, B * 2 * C * N);
  k_rownorm<<<B * C, 256, 0, stream>>>(KVb, NK, 2 * C, 0, C, N);
  k_qk_wmma<<<B * HD, 256, 0, stream>>>(Q, C, KVb, 2 * C, 0, SC, HD, N);
  k_softmax16<<<1, 256, 0, stream>>>(SC, NQ, NK, temp, HD, B * HD * 16, 0);  // q NOT normalized
  k_av_wmma<<<(unsigned)(B * HD * (N / 16)), 32, 0, stream>>>(SC, KVb, 2 * C, C, A, HD, N);

  // ---- 6) x = rgb + conv1x1(out_self * out_cross, w_proj) -> Q ----
  //      (elementwise product fused into the GEMM B-operand load)
  k_gemm1x1_wmma<<<(unsigned)(B * 4 * (N / 16)), 32, 0, stream>>>(X, A, w_proj, Q, rgb, C, C, 4, N);

  // ---- 7) GDFN: y = x + conv1x1(gelu(x1)*x2, w_ffn_out) ----
  k_chan_ln<<<blocks((long long)B * N, TB), TB, 0, stream>>>(Q, ln2_g, ln2_b, A, B, C, N);
  k_gemm1x1_wmma<<<(unsigned)(B * 22 * (N / 16)), 32, 0, stream>>>(A, nullptr, w_fin, F1, nullptr, C, 2 * HID, 22, N);
  k_dw3_gate<<<blocks((long long)B * HID * N, TB), TB, 0, stream>>>(F1, w_fdw, G, HID, H, W, B * HID * N);
  k_gemm1x1_wmma<<<(unsigned)(B * 4 * (N / 16)), 32, 0, stream>>>(G, nullptr, w_fout, y, Q, HID, C, 4, N);
}